// SSTModel_76630806495407
// MI455X (gfx1250) — compile-verified
//
#include <hip/hip_runtime.h>
#include <stdint.h>

// SST forward: x (32, 1048576) f32  ->  sst (32, 524288, 2) f32.
//
// Reformulated scatter(-1/0/+1) as a gather -> no atomics, no zero-init,
// deterministic. Pure HBM-bound (268 MB traffic -> ~11.5 us floor at 23.3 TB/s);
// uses the CDNA5 async global->LDS path for the input tile and 128-bit
// coalesced stores for the output.

#define INV_SQRT2 0.70710678118654752440f

constexpr int BDIM   = 256;
constexpr int TILE_M = 1024;            // output m-values per block
constexpr int TILE_F = 2 * TILE_M;      // input floats per tile (no halo)
constexpr int LDS_F  = TILE_F + 8;      // 4-float halo on each side -> 2056
constexpr int NVEC   = LDS_F / 4;       // 514 b128 transfers per block

__device__ __forceinline__ void async_load_b128_to_lds(uint32_t lds_byte_off,
                                                       uint64_t gaddr) {
    // per-lane async copy of 16 bytes, global -> LDS, tracked by ASYNCcnt
    asm volatile("global_load_async_to_lds_b128 %0, %1, off"
                 :: "v"(lds_byte_off), "v"(gaddr)
                 : "memory");
}

__device__ __forceinline__ void wait_asynccnt0() {
#if __has_builtin(__builtin_amdgcn_s_wait_asynccnt)
    __builtin_amdgcn_s_wait_asynccnt(0);
#else
    asm volatile("s_wait_asynccnt 0" ::: "memory");
#endif
}

__global__ __launch_bounds__(BDIM)
void sst_fwd_kernel(const float* __restrict__ x, float* __restrict__ out,
                    int L, int rowlen) {
    __shared__ __align__(16) float tile[LDS_F];

    const int tid    = threadIdx.x;
    const int b      = blockIdx.y;
    const int m_base = blockIdx.x * TILE_M;

    const float* xrow = x   + (size_t)b * (size_t)rowlen;
    float*       orow = out + (size_t)b * (size_t)rowlen;

    // tile[i] <- xrow[2*m_base - 4 + i], i in [0, LDS_F), edges clamped in-row.
    const int gbase = 2 * m_base - 4;
    for (int v = tid; v < NVEC; v += BDIM) {
        int gidx = gbase + 4 * v;
        gidx = gidx < 0 ? 0 : gidx;
        gidx = gidx > rowlen - 4 ? rowlen - 4 : gidx;
        uint32_t lds_off = (uint32_t)(uintptr_t)(&tile[4 * v]);
        uint64_t gaddr   = (uint64_t)(uintptr_t)(xrow + gidx);
        async_load_b128_to_lds(lds_off, gaddr);
    }
    wait_asynccnt0();
    __syncthreads();

    const int Lm1 = L - 1;

    #pragma unroll
    for (int c = 0; c < 2; ++c) {
        const int s   = c * BDIM + tid;     // float4 store index in [0, 512)
        const int m0  = m_base + 2 * s;     // first of the two m handled
        const int lm0 = 2 * s;              // local m within tile
        float r[4];

        #pragma unroll
        for (int j = 0; j < 2; ++j) {
            const int m = m0 + j;
            const int t = 2 * (lm0 + j) + 4;     // tile index of xe for l = m

            const float xe = tile[t], xo = tile[t + 1];
            const float cA = (xe + xo) * INV_SQRT2;
            const float cD = (xe - xo) * INV_SQRT2;
            const int   d  = (int)(cD < 0.0f) - (int)(cA < 0.0f);

            float acc = 0.0f;
            if (m == 0) {                 // clip(0 + d) == 0 for d <= 0
                if (d <= 0) acc += cA;
            } else if (m == Lm1) {        // clip(L-1 + d) == L-1 for d >= 0
                if (d >= 0) acc += cA;
            } else {
                if (d == 0) acc += cA;
            }
            if (m > 0) {                  // l = m-1 lands here iff d[m-1] == +1
                const float pe = tile[t - 2], po = tile[t - 1];
                const float pA = (pe + po) * INV_SQRT2;
                const float pD = (pe - po) * INV_SQRT2;
                if ((pD < 0.0f) && !(pA < 0.0f)) acc += pA;
            }
            if (m < Lm1) {                // l = m+1 lands here iff d[m+1] == -1
                const float ne = tile[t + 2], no_ = tile[t + 3];
                const float nA = (ne + no_) * INV_SQRT2;
                const float nD = (ne - no_) * INV_SQRT2;
                if (!(nD < 0.0f) && (nA < 0.0f)) acc += nA;
            }
            r[2 * j]     = acc;           // sst[b, m, 0]
            r[2 * j + 1] = cD;            // sst[b, m, 1]  (identity channel)
        }

        *reinterpret_cast<float4*>(orow + 4 * s) =
            make_float4(r[0], r[1], r[2], r[3]);
    }
}

extern "C" void kernel_launch(void* const* d_in, const int* in_sizes, int n_in,
                              void* d_out, int out_size, void* d_ws, size_t ws_size,
                              hipStream_t stream) {
    (void)n_in; (void)out_size; (void)d_ws; (void)ws_size;
    const float* x  = (const float*)d_in[0];
    float*      out = (float*)d_out;

    const int Bn     = 32;
    const int rowlen = in_sizes[0] / Bn;      // N = 1048576
    const int L      = rowlen / 2;            // 524288

    dim3 grid(L / TILE_M, Bn);                // (512, 32)
    sst_fwd_kernel<<<grid, BDIM, 0, stream>>>(x, out, L, rowlen);
}